// NoTradeRegionRNN_14894946583096
// MI455X (gfx1250) — compile-verified
//
#include <hip/hip_runtime.h>
#include <hip/hip_bf16.h>
#include <stdint.h>

// ---------------------------------------------------------------------------
// NoTradeRegionRNN, T=512, B=65536, H=I=1 -> 65536 independent scalar
// recurrences; pure streaming (403 MB => ~17 us floor @ 23.3 TB/s).
//
// CDNA5 pipeline: each wave double-buffers its two input streams through
// wave-private LDS with GLOBAL_LOAD_ASYNC_TO_LDS_B32 (ASYNCcnt path),
// th:TH_LOAD_NT so the touch-once streams don't churn L2. While block k
// (7 steps) is computed from buffer k&1, the async engine fills buffer
// (k+1)&1; one s_wait_asynccnt per 7 steps, no barriers (wave-private
// buffers, per-wave ASYNCcnt, in-order async-load completion).
//
// Addressing split: time offsets are wave-uniform -> carried in SGPR row
// bases (scalar adds, co-issued); the only divergent address component is
// the loop-invariant vaddr = b*4. LDS destination addresses are 28 hoisted
// loop-invariant VGPRs. Keeps VALU count well under the HBM roof.
// ---------------------------------------------------------------------------

#define T_STEPS 512
#define BATCH   65536
#define UNROLL  7           // 511 recurrence steps = 7 * 73, no remainder
#define NBLOCKS 73
#define WAVES_PER_WG 8

__global__ __launch_bounds__(256, 8) void ntr_rnn_async_kernel(
    const float* __restrict__ input,    // [T * B]
    const float* __restrict__ returns,  // [(T-1) * B]
    const float* __restrict__ target,   // [1]  pi_bar
    const float* __restrict__ W_in,     // [1]
    const float* __restrict__ W_h,      // [1]
    const float* __restrict__ b_h,      // [1]
    const float* __restrict__ W_fc1,    // [1]
    const float* __restrict__ W_fc2,    // [1]
    float* __restrict__ out)            // [T * B] output ++ [B] h_final
{
    // Uniform scalar parameters -> s_load_b32 broadcasts.
    const float pi   = target[0];
    const float w_in = W_in[0];
    const float w_h  = W_h[0];
    const float bh   = b_h[0];
    const float w1   = W_fc1[0];
    const float w2   = W_fc2[0];

    const float c_ig  = bh - pi;      // ingate  constant: - (pi - b_h)
    const float c_ig2 = 2.0f * bh;    // ingate2 constant
    const float c_h   = pi + bh;      // h_new   constant

    const int lane = threadIdx.x & 31;
    const int wid  = threadIdx.x >> 5;
    const int b    = blockIdx.x * blockDim.x + threadIdx.x;  // chain per lane

    // Divergent address component, loop-invariant: byte offset of this lane's
    // chain within a row. Row selection lives in uniform SGPR base pointers.
    const uint32_t voff = (uint32_t)b * 4u;

    // Wave-private double buffers: [wave][buf][stream-row][lane]
    // rows 0..6 = input rows t0..t0+6, rows 7..13 = returns rows t0-1..t0+5.
    __shared__ float lbuf[WAVES_PER_WG][2][2 * UNROLL][32];   // 28 KB / WG

    // Issue one 7-step block of async global->LDS copies (14 x b32).
    // GVS mode: mem = SGPR row base + vaddr(b*4); LDS dest = per-row VGPR.
    auto issue_block = [&](int buf, int t0) {
#pragma unroll
        for (int u = 0; u < UNROLL; ++u) {
            const float* gin = input   + (size_t)(t0 + u)     * BATCH; // uniform
            const float* grt = returns + (size_t)(t0 + u - 1) * BATCH; // uniform
            uint32_t lin = (uint32_t)(uintptr_t)&lbuf[wid][buf][u][lane];
            uint32_t lrt = (uint32_t)(uintptr_t)&lbuf[wid][buf][UNROLL + u][lane];
            asm volatile("global_load_async_to_lds_b32 %0, %1, %2 th:TH_LOAD_NT"
                         :: "v"(lin), "v"(voff), "s"(gin) : "memory");
            asm volatile("global_load_async_to_lds_b32 %0, %1, %2 th:TH_LOAD_NT"
                         :: "v"(lrt), "v"(voff), "s"(grt) : "memory");
        }
    };

    // t = 0: h0 = input[0, b]; first output row is a straight copy.
    float h = __builtin_nontemporal_load(&input[b]);
    __builtin_nontemporal_store(h, &out[b]);

    // Prime the pipeline: block 0 (t = 1..7) into buffer 0.
    issue_block(0, 1);

    for (int k = 0; k < NBLOCKS; ++k) {
        const int t0  = 1 + k * UNROLL;
        const int cur = k & 1;

        if (k + 1 < NBLOCKS) {
            // Fill the other buffer one block ahead, then require the 14 ops
            // of the CURRENT block to be done (<=14 outstanding remain).
            issue_block(cur ^ 1, t0 + UNROLL);
            asm volatile("s_wait_asynccnt 14" ::: "memory");
        } else {
            asm volatile("s_wait_asynccnt 0" ::: "memory");
        }

#pragma unroll
        for (int u = 0; u < UNROLL; ++u) {
            float x = lbuf[wid][cur][u][lane];
            float r = lbuf[wid][cur][UNROLL + u][lane];
            // h_adj = h * (1 + r) / (1 + h * r)   (v_rcp_f32: 1+h*r ~ 1)
            float denom = __builtin_fmaf(h, r, 1.0f);
            float h_adj = h * (1.0f + r) * __builtin_amdgcn_rcpf(denom);
            // ingate  = x*w_in + w_h*h_adj - (pi - b_h)
            float ig  = __builtin_fmaf(x, w_in, __builtin_fmaf(w_h, h_adj, c_ig));
            // ingate2 = w_fc1*relu(ingate) + 2*b_h
            float ig2 = __builtin_fmaf(w1, fmaxf(ig, 0.0f), c_ig2);
            // h_new   = w_fc2*relu(ingate2) + pi + b_h
            h = __builtin_fmaf(w2, fmaxf(ig2, 0.0f), c_h);
            // Uniform row base + lane index -> saddr-form NT store.
            float* orow = out + (size_t)(t0 + u) * BATCH;
            __builtin_nontemporal_store(h, &orow[b]);
        }
    }

    // h_final (== output row T-1) appended after the T*B output block.
    __builtin_nontemporal_store(h, &out[(size_t)T_STEPS * BATCH + b]);
}

extern "C" void kernel_launch(void* const* d_in, const int* in_sizes, int n_in,
                              void* d_out, int out_size, void* d_ws, size_t ws_size,
                              hipStream_t stream) {
    // setup_inputs() dict order:
    // 0: input (T,B,I)  1: target (1,)  2: returns (T-1,B,H)  3: hidden (unused)
    // 4: W_in  5: W_h  6: b_h  7: W_fc1  8: W_fc2
    const float* input   = (const float*)d_in[0];
    const float* target  = (const float*)d_in[1];
    const float* returns = (const float*)d_in[2];
    const float* W_in    = (const float*)d_in[4];
    const float* W_h     = (const float*)d_in[5];
    const float* b_h     = (const float*)d_in[6];
    const float* W_fc1   = (const float*)d_in[7];
    const float* W_fc2   = (const float*)d_in[8];
    float* out = (float*)d_out;

    (void)in_sizes; (void)n_in; (void)out_size; (void)d_ws; (void)ws_size;

    dim3 block(256);                 // 8 wave32 per workgroup
    dim3 grid(BATCH / 256);          // 256 workgroups, 2048 waves total
    ntr_rnn_async_kernel<<<grid, block, 0, stream>>>(
        input, returns, target, W_in, W_h, b_h, W_fc1, W_fc2, out);
}